// RoIHeads_19009525252587
// MI455X (gfx1250) — compile-verified
//
#include <hip/hip_runtime.h>
#include <math.h>

// ---------------- problem constants (match reference) ----------------
#define BB 2
#define PP 512
#define CC 11
#define NN 5120           // P*(C-1)
#define DETS 100
#define MASKW 28
#define MASKPIX (MASKW*MASKW)     // 784
#define IMGW 800.0f
#define SCORE_THRESH 0.05f
#define MIN_SIZE 0.01f
#define NMS_T 0.5f
#define BBOX_CLIP 4.135166556742356f   // log(1000/16)
#define TPD 320           // tiles per dim = NN/16
#define TRI (TPD*(TPD+1)/2)            // 51360 upper-triangle tiles per image
#define W16 320           // uint16 mask words per row
#define W32 160           // u32 mask words per row

typedef __attribute__((ext_vector_type(2))) float v2f;
typedef __attribute__((ext_vector_type(8))) float v8f;
typedef __attribute__((ext_vector_type(4))) unsigned v4u;
typedef __attribute__((ext_vector_type(8))) int v8i;
typedef __attribute__((ext_vector_type(4))) int v4i;

// ---------------- K0: init atomics ----------------
__global__ void k0_init(unsigned* maxbox) {
    if (threadIdx.x < BB) maxbox[threadIdx.x] = 0u;
}

// ---------------- K1: softmax + decode + validity ----------------
__global__ void k1_decode(const float* __restrict__ logits,
                          const float* __restrict__ regr,
                          const float* __restrict__ props,
                          float* __restrict__ eff,       // [B][N]
                          float* __restrict__ decb,      // [B][N][4]
                          unsigned* __restrict__ maxbox) {
    int t = blockIdx.x * blockDim.x + threadIdx.x;     // 0..1023
    if (t >= BB * PP) return;
    int b = t >> 9, p = t & 511;

    // softmax over 11 classes
    float lg[CC];
    float mx = -INFINITY;
    for (int c = 0; c < CC; ++c) { lg[c] = logits[t * CC + c]; mx = fmaxf(mx, lg[c]); }
    float den = 0.f;
    for (int c = 0; c < CC; ++c) { lg[c] = expf(lg[c] - mx); den += lg[c]; }
    float inv = 1.0f / den;

    float px1 = props[t * 4 + 0], py1 = props[t * 4 + 1];
    float px2 = props[t * 4 + 2], py2 = props[t * 4 + 3];
    float w = px2 - px1, h = py2 - py1;
    float cx = px1 + 0.5f * w, cy = py1 + 0.5f * h;

    float localmax = 0.f;
    for (int c = 1; c < CC; ++c) {
        const float* r = regr + t * (4 * CC) + c * 4;
        float dx = r[0] / 10.0f, dy = r[1] / 10.0f;
        float dw = fminf(r[2] / 5.0f, BBOX_CLIP);
        float dh = fminf(r[3] / 5.0f, BBOX_CLIP);
        float pcx = dx * w + cx, pcy = dy * h + cy;
        float pw = expf(dw) * w, ph = expf(dh) * h;
        float x1 = fminf(fmaxf(pcx - 0.5f * pw, 0.f), IMGW);
        float y1 = fminf(fmaxf(pcy - 0.5f * ph, 0.f), IMGW);
        float x2 = fminf(fmaxf(pcx + 0.5f * pw, 0.f), IMGW);
        float y2 = fminf(fmaxf(pcy + 0.5f * ph, 0.f), IMGW);
        int n = p * (CC - 1) + (c - 1);
        float4* d4 = (float4*)decb;
        d4[b * NN + n] = make_float4(x1, y1, x2, y2);
        float sc = lg[c] * inv;
        bool valid = (sc > SCORE_THRESH) && ((x2 - x1) >= MIN_SIZE) && ((y2 - y1) >= MIN_SIZE);
        eff[b * NN + n] = valid ? sc : -INFINITY;
        localmax = fmaxf(localmax, fmaxf(fmaxf(x1, y1), fmaxf(x2, y2)));
    }
    atomicMax(&maxbox[b], __float_as_uint(localmax));
}

// ---------------- K2: per-image bitonic sort + gather ----------------
// TDM stages the 20KB score vector into LDS (one wave issues TENSOR_LOAD_TO_LDS).
__global__ __launch_bounds__(1024) void k2_sort(const float* __restrict__ eff,
                        const float* __restrict__ decb,
                        const unsigned* __restrict__ maxbox,
                        float* __restrict__ sscore,   // [B][N]
                        float* __restrict__ sbox,     // [B][N][4] sorted (un-offset)
                        float* __restrict__ obox,     // [B][N][4] sorted + class offset
                        float* __restrict__ sarea,    // [B][N]
                        int*   __restrict__ slab) {   // [B][N]
    __shared__ float kk[8192];
    __shared__ int   vv[8192];
    int img = blockIdx.x, tid = threadIdx.x;

#if __has_builtin(__builtin_amdgcn_tensor_load_to_lds) && __has_builtin(__builtin_amdgcn_s_wait_tensorcnt)
#define K2_TDM 1
    if (tid < 32) {   // one wave issues the DMA (TDM ignores EXEC; per-wave op)
        unsigned ldsa = (unsigned)(size_t)(void*)&kk[0];        // low 32 bits of flat LDS addr = LDS byte offset
        unsigned long long ga = (unsigned long long)(const void*)(eff + (size_t)img * NN);
        // D# group 0: count=1, lds_addr, 57-bit global_addr, type=2 ("image")
        v4u g0 = { 1u,
                   ldsa,
                   (unsigned)(ga & 0xffffffffull),
                   (unsigned)((ga >> 32) & 0x1ffffffull) | (2u << 30) };
        // D# group 1: data_size=4B(code 2), tensor_dim0=tile_dim0=5120, tensor_dim1=tile_dim1=1
        v8i g1 = { (int)(2u << 16),                 // mask=0 | data_size=2
                   (int)((unsigned)(NN & 0xffff) << 16),   // tensor_dim0[15:0] at bits 63:48
                   (int)(((unsigned)NN >> 16) | (1u << 16)),// tensor_dim0[31:16] | tensor_dim1[15:0]=1
                   (int)((unsigned)NN << 16),       // tensor_dim1[31:16]=0 | tile_dim0=5120
                   1,                               // tile_dim1=1, tile_dim2=0
                   NN,                              // tensor_dim0_stride[31:0]
                   0, 0 };
        v4i gz4 = { 0, 0, 0, 0 };
        v8i gz8 = { 0, 0, 0, 0, 0, 0, 0, 0 };
        __builtin_amdgcn_tensor_load_to_lds(g0, g1, gz4, gz4, gz8, 0);
        __builtin_amdgcn_s_wait_tensorcnt(0);
    }
#endif

    for (int i = tid; i < 8192; i += 1024) {
        vv[i] = i;
#ifdef K2_TDM
        if (i >= NN) kk[i] = -INFINITY;             // pad; [0,NN) filled by TDM
#else
        kk[i] = (i < NN) ? eff[img * NN + i] : -INFINITY;
#endif
    }
    __syncthreads();

    for (int k = 2; k <= 8192; k <<= 1) {
        for (int j = k >> 1; j > 0; j >>= 1) {
            for (int idx = tid; idx < 8192; idx += 1024) {
                int ixj = idx ^ j;
                if (ixj > idx) {
                    float ka = kk[idx], kb = kk[ixj];
                    int   va = vv[idx], vb = vv[ixj];
                    // "a after b" in descending-stable order
                    bool a_lp = (ka < kb) || (ka == kb && va > vb);
                    bool b_lp = (kb < ka) || (kb == ka && vb > va);
                    bool up = (idx & k) == 0;
                    if (up ? a_lp : b_lp) {
                        kk[idx] = kb; kk[ixj] = ka;
                        vv[idx] = vb; vv[ixj] = va;
                    }
                }
            }
            __syncthreads();
        }
    }

    float mb = __uint_as_float(maxbox[img]) + 1.0f;
    const float4* d4 = (const float4*)decb;
    float4* sb4 = (float4*)sbox;
    float4* ob4 = (float4*)obox;
    for (int s = tid; s < NN; s += 1024) {
        int oi = vv[s];
        float sc = kk[s];
        sscore[img * NN + s] = sc;
        int lab = (oi % (CC - 1)) + 1;
        slab[img * NN + s] = lab;
        float4 bx = d4[img * NN + oi];
        sb4[img * NN + s] = bx;
        float off = (float)lab * mb;
        float4 ob = make_float4(bx.x + off, bx.y + off, bx.z + off, bx.w + off);
        ob4[img * NN + s] = ob;
        sarea[img * NN + s] = (ob.z - ob.x) * (ob.w - ob.y);
    }
}

// ---------------- K3: pairwise IoU bitmask, WMMA for area outer-sum ----------------
// Upper-triangle tiles only, enumerated exactly (sqrt decode + fixup).
__global__ __launch_bounds__(256) void k3_ioumask(const float* __restrict__ obox,
                           const float* __restrict__ sarea,
                           unsigned short* __restrict__ m16) {  // [B][N][320]
    int wv = blockIdx.x * 8 + (threadIdx.x >> 5);     // global wave id
    if (wv >= BB * TRI) return;
    int lane = threadIdx.x & 31;
    int img = wv / TRI;
    int u   = wv % TRI;
    // decode u -> (ti, tj), tj >= ti, counting from the bottom-right corner
    int vr = TRI - 1 - u;
    int k = (int)((sqrtf(8.0f * (float)vr + 1.0f) - 1.0f) * 0.5f);
    while ((k + 1) * (k + 2) / 2 <= vr) ++k;
    while (k * (k + 1) / 2 > vr) --k;
    int o = vr - k * (k + 1) / 2;
    int ti = TPD - 1 - k;
    int tj = TPD - 1 - o;

    const float* ar = sarea + img * NN;
    // A (16x4): row M=lane (lanes 0-15): K0=area_i, K1=1 ; lanes 16-31 carry K2,K3 = 0
    // B (4x16): col N=lane (lanes 0-15): K0=1, K1=area_j ; lanes 16-31 carry K2,K3 = 0
    float ai = (lane < 16) ? ar[ti * 16 + lane] : 0.f;
    float aj = (lane < 16) ? ar[tj * 16 + lane] : 0.f;
    v2f A;  A.x  = ai;                       A.y  = (lane < 16) ? 1.f : 0.f;
    v2f Bm; Bm.x = (lane < 16) ? 1.f : 0.f;  Bm.y = aj;
    v8f Cz = {0.f, 0.f, 0.f, 0.f, 0.f, 0.f, 0.f, 0.f};
    v8f D = __builtin_amdgcn_wmma_f32_16x16x4_f32(false, A, false, Bm,
                                                  (short)0, Cz, false, false);

    const float4* ob4 = (const float4*)obox;
    const float4* bip = ob4 + (size_t)img * NN + (size_t)ti * 16;
    float4 bj = ob4[(size_t)img * NN + (size_t)tj * 16 + (lane & 15)];
    int mbase = (lane >= 16) ? 8 : 0;
    unsigned hw = 0u;   // per-lane halfword for row ti*16+lane (lanes 0-15)
    #pragma unroll
    for (int v = 0; v < 8; ++v) {
        int M = v + mbase;
        float4 bi = bip[M];
        float ix = fmaxf(fminf(bi.z, bj.z) - fmaxf(bi.x, bj.x), 0.f);
        float iy = fmaxf(fminf(bi.w, bj.w) - fmaxf(bi.y, bj.y), 0.f);
        float inter = ix * iy;
        float uni = fmaxf(D[v] - inter, 1e-6f);
        bool pred = inter > NMS_T * uni;               // iou > 0.5
        unsigned bal = (unsigned)__ballot(pred);
        // ballot bits 0..15: row ti*16+v ; bits 16..31: row ti*16+v+8 (cols tj*16..+15)
        hw = (lane == v)     ? (bal & 0xffffu) : hw;   // branch-free selects
        hw = (lane == v + 8) ? (bal >> 16)     : hw;
    }
    if (lane < 16)
        m16[(size_t)(img * NN + ti * 16 + lane) * W16 + tj] = (unsigned short)hw;
}

// ---------------- K4: one-wave sequential greedy sweep ----------------
__global__ void k4_sweep(const unsigned short* __restrict__ m16,
                         const float* __restrict__ sscore,
                         int* __restrict__ kept) {     // [B][DETS]
    int img = blockIdx.x, lane = threadIdx.x;          // 32 threads
    const unsigned* rowbase = (const unsigned*)(m16 + (size_t)img * NN * W16);
    const float* ss = sscore + img * NN;
    int* kp = kept + img * DETS;
    unsigned r[5] = {0u, 0u, 0u, 0u, 0u};              // lane l owns words q*32+l
    int cnt = 0;
    #pragma unroll
    for (int s = 0; s < 5; ++s) {
        for (int i2 = 0; i2 < 1024; ++i2) {
            int i = (s << 10) + i2;
            int w = i >> 5;                            // word index; w>>5 == s here
            unsigned word = __shfl(r[s], w & 31, 32);
            bool rem = (word >> (i & 31)) & 1u;
            float sc = ss[i];
            if (!rem && sc > -1e37f) {
                const unsigned* rp = rowbase + (size_t)i * W32;
                #pragma unroll
                for (int q = 0; q < 5; ++q) r[q] |= rp[q * 32 + lane];
                if (lane == 0) kp[cnt] = i;
                ++cnt;
                if (cnt >= DETS) goto done;
            }
        }
    }
done:
    if (lane == 0) for (int d = cnt; d < DETS; ++d) kp[d] = -1;
}

// ---------------- K5: outputs (boxes/scores/labels + mask sigmoid gather) ----------------
__global__ void k5_out(const int* __restrict__ kept,
                       const float* __restrict__ sbox,
                       const float* __restrict__ sscore,
                       const int* __restrict__ slab,
                       const float* __restrict__ mlogits,
                       float* __restrict__ out) {
    int bd = blockIdx.x;                    // 0..199
    int b = bd / DETS, d = bd % DETS;
    int ki = kept[b * DETS + d];
    int lab = 0;
    float osc = 0.f;
    float4 obx = make_float4(0.f, 0.f, 0.f, 0.f);
    if (ki >= 0) {
        lab = slab[b * NN + ki];
        osc = sscore[b * NN + ki];
        obx = ((const float4*)sbox)[b * NN + ki];
    }
    if (threadIdx.x == 0) {
        out[bd * 4 + 0] = obx.x; out[bd * 4 + 1] = obx.y;
        out[bd * 4 + 2] = obx.z; out[bd * 4 + 3] = obx.w;
        out[800 + bd] = osc;
        out[1000 + bd] = (float)lab;
    }
    const float* src = mlogits + ((size_t)(b * DETS + d) * CC + lab) * MASKPIX;
    float* dst = out + 1200 + (size_t)bd * MASKPIX;
    for (int pix = threadIdx.x; pix < MASKPIX; pix += blockDim.x) {
        float x = src[pix];
        dst[pix] = 1.0f / (1.0f + expf(-x));
    }
}

// ---------------- launcher ----------------
extern "C" void kernel_launch(void* const* d_in, const int* in_sizes, int n_in,
                              void* d_out, int out_size, void* d_ws, size_t ws_size,
                              hipStream_t stream) {
    const float* logits = (const float*)d_in[0];   // (1024, 11)
    const float* regr   = (const float*)d_in[1];   // (1024, 44)
    const float* props  = (const float*)d_in[2];   // (2, 512, 4)
    const float* mlog   = (const float*)d_in[3];   // (2, 100, 11, 28, 28)
    float* out = (float*)d_out;

    char* p = (char*)d_ws;
    auto take = [&](size_t bytes) { char* r = p; p += (bytes + 255) & ~size_t(255); return r; };
    unsigned* maxbox = (unsigned*)take(BB * sizeof(unsigned));
    float* eff    = (float*)take((size_t)BB * NN * sizeof(float));
    float* decb   = (float*)take((size_t)BB * NN * 4 * sizeof(float));
    float* sscore = (float*)take((size_t)BB * NN * sizeof(float));
    float* sbox   = (float*)take((size_t)BB * NN * 4 * sizeof(float));
    float* obox   = (float*)take((size_t)BB * NN * 4 * sizeof(float));
    float* sarea  = (float*)take((size_t)BB * NN * sizeof(float));
    int*   slab   = (int*)take((size_t)BB * NN * sizeof(int));
    int*   kept   = (int*)take((size_t)BB * DETS * sizeof(int));
    unsigned short* m16 = (unsigned short*)take((size_t)BB * NN * W16 * sizeof(unsigned short));
    (void)ws_size; (void)n_in; (void)in_sizes; (void)out_size;

    k0_init<<<1, 64, 0, stream>>>(maxbox);
    k1_decode<<<2, 512, 0, stream>>>(logits, regr, props, eff, decb, maxbox);
    k2_sort<<<BB, 1024, 0, stream>>>(eff, decb, maxbox, sscore, sbox, obox, sarea, slab);
    k3_ioumask<<<(BB * TRI + 7) / 8, 256, 0, stream>>>(obox, sarea, m16);
    k4_sweep<<<BB, 32, 0, stream>>>(m16, sscore, kept);
    k5_out<<<BB * DETS, 256, 0, stream>>>(kept, sbox, sscore, slab, mlog, out);
}